// TwoChannel_Conv_TraceLIF_MinWTA_17059610100018
// MI455X (gfx1250) — compile-verified
//
#include <hip/hip_runtime.h>

// Problem constants
#define BATCH 128
#define LX    4096
#define LP    4088   // 4096 - (KMAX-1)
#define KF    32     // 4 groups x 8 filters

static constexpr size_t NEL  = (size_t)BATCH * LP * KF;            // 16,744,448 per array
static constexpr size_t CHAN = 6 * NEL + BATCH + (size_t)BATCH * KF + 1; // per-channel slab

typedef __attribute__((ext_vector_type(2))) float v2f;
typedef __attribute__((ext_vector_type(8))) float v8f;

// Build the width-9 zero-left-padded causal filter for feature f (group g -> k=3+2g).
__device__ __forceinline__ void make_wp(int f,
    const float* __restrict__ w3, const float* __restrict__ b3,
    const float* __restrict__ w5, const float* __restrict__ b5,
    const float* __restrict__ w7, const float* __restrict__ b7,
    const float* __restrict__ w9, const float* __restrict__ b9,
    float Wp[9], float& bias)
{
    const int g = f >> 3, c = f & 7;
    const int k = 3 + 2 * g;
    const float* wk = (g == 0) ? w3 : (g == 1) ? w5 : (g == 2) ? w7 : w9;
    const float* bk = (g == 0) ? b3 : (g == 1) ? b5 : (g == 2) ? b7 : b9;
#pragma unroll
    for (int j = 0; j < 9; ++j) Wp[j] = 0.0f;
    for (int i = 0; i < k; ++i) Wp[9 - k + i] = wk[c * k + i];
    bias = bk[c];
}

// ---------------------------------------------------------------------------
// Kernel 1: filterbank as WMMA GEMM. One wave = one 16(t) x 16(f) tile of d2.
// d2[b, i, f] = relu( sum_{j=0..8} Wp[f][j] * x[b, i + j] + bias[f] ), i in [0,4088)
// K=9 padded to 12 -> 3x V_WMMA_F32_16X16X4_F32, accumulator preloaded with bias.
// d2 is write-once -> non-temporal stores (keep 192MB L2 for the s array).
// ---------------------------------------------------------------------------
__global__ __launch_bounds__(32) void conv_wmma(
    const float* __restrict__ x,
    const float* __restrict__ w3, const float* __restrict__ b3,
    const float* __restrict__ w5, const float* __restrict__ b5,
    const float* __restrict__ w7, const float* __restrict__ b7,
    const float* __restrict__ w9, const float* __restrict__ b9,
    float* __restrict__ d2)
{
    const int l  = threadIdx.x;          // lane 0..31
    const int i0 = blockIdx.x * 16;      // time tile (256 tiles cover 0..4095, clipped at 4088)
    const int b  = blockIdx.y;           // batch row
    const int f0 = blockIdx.z * 16;      // feature tile (0 or 16)
    const int n  = l & 15;               // matrix column / A row (per ISA layout)
    const int kh = l >> 4;               // K-half select

    float Wp[9]; float bias;
    make_wp(f0 + n, w3, b3, w5, b5, w7, b7, w9, b9, Wp, bias);

    const float* xr = x + (size_t)b * LX;

    // A: 16x4 f32 per chunk. lane<16: VGPR0=K0,VGPR1=K1; lane>=16: K2,K3. Row m = l&15.
    auto ldx = [&](int j) -> float {
        const int xi = i0 + n + j;
        return (xi < LX) ? xr[xi] : 0.0f;   // guards only rows >= LP (never stored)
    };
    auto wj = [&](int j) -> float { return (j < 9) ? Wp[j] : 0.0f; };

    v2f a0, a1, a2, bm0, bm1, bm2;
    a0.x = ldx(0 + 2 * kh);  a0.y = ldx(1 + 2 * kh);
    a1.x = ldx(4 + 2 * kh);  a1.y = ldx(5 + 2 * kh);
    a2.x = (kh == 0) ? ldx(8) : 0.0f;  a2.y = 0.0f;     // j=9..11 are K-padding

    // B: 4x16 f32 per chunk. VGPR0=row(4c+2kh), VGPR1=row(4c+2kh+1), col n = l&15.
    bm0.x = wj(0 + 2 * kh);  bm0.y = wj(1 + 2 * kh);
    bm1.x = wj(4 + 2 * kh);  bm1.y = wj(5 + 2 * kh);
    bm2.x = wj(8 + 2 * kh);  bm2.y = wj(9 + 2 * kh);    // rows 9..11 are zero

    v8f c;
#pragma unroll
    for (int r = 0; r < 8; ++r) c[r] = bias;            // bias broadcast down each column

    c = __builtin_amdgcn_wmma_f32_16x16x4_f32(false, a0, false, bm0, (short)0, c, false, false);
    c = __builtin_amdgcn_wmma_f32_16x16x4_f32(false, a1, false, bm1, (short)0, c, false, false);
    c = __builtin_amdgcn_wmma_f32_16x16x4_f32(false, a2, false, bm2, (short)0, c, false, false);

    // D layout: VGPR r -> row (r + 8*kh), col n. Clip tail rows >= LP.
    const size_t rowbase = (size_t)b * LP * KF + (size_t)(f0 + n);
#pragma unroll
    for (int r = 0; r < 8; ++r) {
        const int i = i0 + r + 8 * kh;
        if (i < LP)
            __builtin_nontemporal_store(fmaxf(c[r], 0.0f), &d2[rowbase + (size_t)i * KF]);
    }
}

// ---------------------------------------------------------------------------
// Kernel 2: trace + LIF scan + WTA. One wave = one batch row, lane = feature.
// Writes z, s, series, v per step; then wave-reduction WTA; then s_g pass.
// z/series/v/s_g are write-once -> non-temporal; s is re-read (s_g pass) ->
// regular temporal so the 2x64MB s arrays stay resident in the 192MB L2.
// ---------------------------------------------------------------------------
__global__ __launch_bounds__(256) void scan_lif(
    const float* __restrict__ x,
    const float* __restrict__ w3, const float* __restrict__ b3,
    const float* __restrict__ w5, const float* __restrict__ b5,
    const float* __restrict__ w7, const float* __restrict__ b7,
    const float* __restrict__ w9, const float* __restrict__ b9,
    float* out)
{
    const int tid = threadIdx.x;
    const int f = tid & 31;
    const int b = blockIdx.x * 8 + (tid >> 5);

    float Wp[9]; float bias;
    make_wp(f, w3, b3, w5, b5, w7, b7, w9, b9, Wp, bias);

    const float* xr = x + (size_t)b * LX;
    float xw[9];
#pragma unroll
    for (int j = 0; j < 9; ++j) xw[j] = xr[j];          // broadcast loads (uniform per wave)

    float* zo   = out + NEL;
    float* so   = out + 2 * NEL;
    float* sero = out + 3 * NEL;
    float* vo   = out + 4 * NEL;
    const size_t base = (size_t)b * LP * KF + f;

    float z = 0.0f, v = 0.0f, a = 0.0f, snum = 0.0f;
    int tfi = LP;                                        // first-spike step, LP if none
    const float invTau = 1.0f / 20.0f, invTauA = 1.0f / 100.0f;

    for (int i = 0; i < LP; ++i) {
        // filterbank (f32, same math as WMMA kernel up to summation order)
        float d = bias;
#pragma unroll
        for (int j = 0; j < 9; ++j) d = fmaf(Wp[j], xw[j], d);
        d = fmaxf(d, 0.0f);
#pragma unroll
        for (int j = 0; j < 8; ++j) xw[j] = xw[j + 1];
        if (i + 9 < LX) xw[8] = xr[i + 9];

        // trace
        z = 0.95f * z + 0.05f * d;

        // LIF (forward values of the surrogate-gradient formulation)
        const float th = 1.5f + 1.5f * a;
        v = v + (z - v * invTau);
        const float vout = v;
        const float u = 10.0f * (v - th);
        const float ssoft = 1.0f / (1.0f + __expf(-u));
        const float shard = (v >= th) ? 1.0f : 0.0f;
        const float s = (shard - ssoft) + ssoft;        // forward == shard (with ref rounding)
        snum += s;
        v = (shard != 0.0f) ? -0.5f : v;                // exact reset select
        a = a + (s - a * invTauA);
        tfi = (shard != 0.0f) ? min(tfi, i) : tfi;

        const size_t o = base + (size_t)i * KF;
        __builtin_nontemporal_store(z,    &zo[o]);      // write-only streams: NT
        __builtin_nontemporal_store(snum, &sero[o]);
        __builtin_nontemporal_store(vout, &vo[o]);
        so[o] = s;                                      // re-read below: keep temporal
    }

    // ---- WTA over the 32 lanes of this wave ----
    // min over (t_first, f): first spiking time, smallest feature on ties
    int key = (tfi << 5) | f;
#pragma unroll
    for (int off = 16; off > 0; off >>= 1) key = min(key, __shfl_xor(key, off, 32));
    const int tstar = key >> 5;
    const int kstar = key & 31;
    const bool hasAny = (tstar < LP);

    // fallback: argmax of spike count, first index on ties
    float cnt = snum; int ci = f;
#pragma unroll
    for (int off = 16; off > 0; off >>= 1) {
        const float oc = __shfl_xor(cnt, off, 32);
        const int   oi = __shfl_xor(ci,  off, 32);
        if (oc > cnt || (oc == cnt && oi < ci)) { cnt = oc; ci = oi; }
    }
    const int idx = hasAny ? kstar : ci;

    // w_sur = softmax(-t_first / 0.2) (max-subtracted, matches ref incl. no-spike uniform case)
    const float e = __expf(-5.0f * (float)(tfi - tstar));
    float ssum = e;
#pragma unroll
    for (int off = 16; off > 0; off >>= 1) ssum += __shfl_xor(ssum, off, 32);
    const float wsur  = e / ssum;
    const float whard = (f == idx) ? 1.0f : 0.0f;
    const float wv    = (whard - wsur) + wsur;          // forward == one-hot (with ref rounding)

    out[5 * NEL + BATCH + (size_t)b * KF + f] = wv;
    if (f == 0) out[5 * NEL + b] = (float)idx;
    if (b == 0 && f == 0) out[6 * NEL + BATCH + (size_t)BATCH * KF] = 8.0f;  // warmup

    // ---- s_g = s * w : only the winning lane reads s back (L2-resident) ----
    float* sg = out + 5 * NEL + BATCH + (size_t)BATCH * KF;
    const float* sin = out + 2 * NEL;
    for (int i = 0; i < LP; ++i) {
        const size_t o = base + (size_t)i * KF;
        float val = 0.0f;
        if (wv != 0.0f) val = sin[o] * wv;
        __builtin_nontemporal_store(val, &sg[o]);       // write-only: NT
    }
}

// ---------------------------------------------------------------------------
// Host launch. Inputs: x, y, then 8 param leaves per channel. Handles both
// insertion-order (w3,b3,...) and sorted-key (b3..b9,w3..w9) flattening by
// inspecting in_sizes (w3 has 24 elements, b3 has 8).
// ---------------------------------------------------------------------------
extern "C" void kernel_launch(void* const* d_in, const int* in_sizes, int n_in,
                              void* d_out, int out_size, void* d_ws, size_t ws_size,
                              hipStream_t stream)
{
    (void)d_ws; (void)ws_size; (void)out_size;
    if (n_in < 18) return;

    const float* x = (const float*)d_in[0];
    const float* y = (const float*)d_in[1];
    const float* px[8]; const float* py[8];   // canonical: w3,b3,w5,b5,w7,b7,w9,b9

    if (in_sizes[2] == 24) {                  // insertion order
        for (int i = 0; i < 8; ++i) { px[i] = (const float*)d_in[2 + i]; py[i] = (const float*)d_in[10 + i]; }
    } else {                                  // sorted keys: b3,b5,b7,b9,w3,w5,w7,w9
        const int wmap[4] = {6, 7, 8, 9}, bmap[4] = {2, 3, 4, 5};
        for (int g = 0; g < 4; ++g) {
            px[2 * g]     = (const float*)d_in[wmap[g]];
            px[2 * g + 1] = (const float*)d_in[bmap[g]];
            py[2 * g]     = (const float*)d_in[wmap[g] + 8];
            py[2 * g + 1] = (const float*)d_in[bmap[g] + 8];
        }
    }

    float* out  = (float*)d_out;
    float* outx = out;
    float* outy = out + CHAN;

    const dim3 cgrid(256, 128, 2), cblk(32);
    conv_wmma<<<cgrid, cblk, 0, stream>>>(x, px[0], px[1], px[2], px[3], px[4], px[5], px[6], px[7], outx);
    conv_wmma<<<cgrid, cblk, 0, stream>>>(y, py[0], py[1], py[2], py[3], py[4], py[5], py[6], py[7], outy);

    const dim3 sgrid(16), sblk(256);
    scan_lif<<<sgrid, sblk, 0, stream>>>(x, px[0], px[1], px[2], px[3], px[4], px[5], px[6], px[7], outx);
    scan_lif<<<sgrid, sblk, 0, stream>>>(y, py[0], py[1], py[2], py[3], py[4], py[5], py[6], py[7], outy);
}